// GHMCLoss_69793218560424
// MI455X (gfx1250) — compile-verified
//
#include <hip/hip_runtime.h>
#include <stdint.h>

#define NROWS   4194304
#define NCLS    8
#define NBINS   10
#define TROWS   512
#define TELEMS  (TROWS * NCLS)      // 4096 floats / pred tile (16 KB)
#define BLOCK   256                 // 8 wave32s; 2 rows per thread per tile
#define NBLK    2048
#define NTILES  (NROWS / TROWS)     // 8192
#define TPB     (NTILES / NBLK)     // 4 tiles per block

typedef __attribute__((ext_vector_type(4))) unsigned int v4u;
typedef __attribute__((ext_vector_type(8))) int          v8i;
typedef __attribute__((ext_vector_type(4))) int          v4i;

__device__ __forceinline__ unsigned rfl(unsigned x) {
  return (unsigned)__builtin_amdgcn_readfirstlane((int)x);
}

// Issue a 1-D TDM DMA: `elems` 4-byte elements from global -> LDS.
// D# group0: count=1, lds_addr, 57-bit global addr, type=2 ("image").
// D# group1: data_size=4B (code 2), tensor_dim0=tile_dim0=elems,
//            tensor_dim1=1, workgroup_mask=0 (not in a cluster).
__device__ __forceinline__ void tdm_load_1d(unsigned ldsAddr, const void* gptr,
                                            unsigned elems) {
  uint64_t ga = (uint64_t)(uintptr_t)gptr;
  v4u g0;
  g0[0] = rfl(1u);                                              // count=1
  g0[1] = rfl(ldsAddr);                                         // lds_addr
  g0[2] = rfl((unsigned)ga);                                    // gaddr[31:0]
  g0[3] = rfl(((unsigned)(ga >> 32) & 0x01FFFFFFu) | 0x80000000u); // gaddr[56:32] | type=2
  v8i g1;
  g1[0] = (int)rfl(0x00020000u);                 // data_size = 4B
  g1[1] = (int)rfl((elems & 0xFFFFu) << 16);     // tensor_dim0[15:0] @ bits 63:48
  g1[2] = (int)rfl((elems >> 16) | (1u << 16));  // tensor_dim0[31:16] | tensor_dim1=1
  g1[3] = (int)rfl((elems & 0xFFFFu) << 16);     // tile_dim0 @ bits 127:112
  g1[4] = 0;                                     // tile_dim1=0, tile_dim2=0
  g1[5] = (int)rfl(elems);                       // tensor_dim0_stride (unused, 1-D)
  g1[6] = 0;
  g1[7] = 0;
  v4i z4 = {0, 0, 0, 0};
  v8i z8 = {0, 0, 0, 0, 0, 0, 0, 0};
  __builtin_amdgcn_tensor_load_to_lds(g0, g1, z4, z4, z8, 0);  // clang-23 6-arg form
}

__global__ void ghm_zero(unsigned* gCnt, float* gSum, float* gW) {
  const int t = threadIdx.x;
  if (t < NBINS) { gCnt[t] = 0u; gSum[t] = 0.0f; }
  if (t == 0) gW[0] = 0.0f;
}

__global__ __launch_bounds__(BLOCK) void ghm_pass1(
    const float* __restrict__ pred, const int* __restrict__ tgt,
    const float* __restrict__ weight,
    unsigned* __restrict__ gCnt, float* __restrict__ gSum,
    float* __restrict__ gW) {
  __shared__ float    sPred[2][TELEMS];    // 2 x 16 KB, TDM double buffer
  __shared__ int      sTgt[2][TROWS];      // 2 x  2 KB
  __shared__ float    sSum[NBINS][BLOCK];  // per-thread slots, bank = tid%64
  __shared__ unsigned sCnt[NBINS][BLOCK];  // per-thread slots, bank = tid%64
  __shared__ float    sWred[BLOCK];
  __shared__ float    wtab[NCLS];

  const int tid = threadIdx.x;
  if (tid < NCLS) wtab[tid] = weight[tid];
#pragma unroll
  for (int b = 0; b < NBINS; ++b) { sSum[b][tid] = 0.0f; sCnt[b][tid] = 0u; }

  const int tile0 = blockIdx.x * TPB;
  float wacc = 0.0f;   // one weight per ROW (x8 at the end)

  // Prime the pipeline: wave 0 issues the first tile's two DMAs.
  if (tid < 32) {
    tdm_load_1d((unsigned)(uintptr_t)&sPred[0][0],
                pred + (size_t)tile0 * TELEMS, TELEMS);
    tdm_load_1d((unsigned)(uintptr_t)&sTgt[0][0],
                tgt + (size_t)tile0 * TROWS, TROWS);
  }

  for (int t = 0; t < TPB; ++t) {
    const int cur = t & 1, nxt = cur ^ 1;
    if (tid < 32) __builtin_amdgcn_s_wait_tensorcnt(0); // tile t landed in LDS
    __syncthreads(); // data visible; everyone done reading buf[nxt]
    if (t + 1 < TPB && tid < 32) {
      const int g = tile0 + t + 1;
      tdm_load_1d((unsigned)(uintptr_t)&sPred[nxt][0],
                  pred + (size_t)g * TELEMS, TELEMS);
      tdm_load_1d((unsigned)(uintptr_t)&sTgt[nxt][0],
                  tgt + (size_t)g * TROWS, TROWS);
    }
    const float4* tp4 = (const float4*)&sPred[cur][0];
    const int*    tt  = sTgt[cur];

#pragma unroll
    for (int k = 0; k < TROWS / BLOCK; ++k) {    // 2 rows per thread
      const int   r  = tid + k * BLOCK;          // conflict-free row fetches
      const int   tc = tt[r];
      const float wv = wtab[tc];                 // <=8 addrs, distinct banks
      const float wvln2 = wv * 0.6931471806f;    // hoisted per row
      const float4 p0 = tp4[2 * r + 0];          // ds_load_b128
      const float4 p1 = tp4[2 * r + 1];
      const float xs[8] = {p0.x, p0.y, p0.z, p0.w, p1.x, p1.y, p1.z, p1.w};
      wacc += wv;
#pragma unroll
      for (int c = 0; c < NCLS; ++c) {
        const float x  = xs[c];
        // u = x if y==0 else -x  =>  g = sigmoid(u), bce = softplus(u)
        const float u  = (tc == c) ? -x : x;
        const float au = fabsf(u);
        const float em = __builtin_amdgcn_exp2f(au * -1.442695041f); // e^{-|u|}
        const float d  = 1.0f + em;
        const float inv = __builtin_amdgcn_rcpf(d);                  // 1/(1+em)
        const float g  = (u >= 0.0f) ? inv : (1.0f - inv);           // sigmoid(u)
        int b = (int)(g * 10.0f);
        b = (b > NBINS - 1) ? (NBINS - 1) : b;
        // wv*softplus(u) = fma(log2(d), wv*ln2, max(u,0)*wv)
        const float bcew = fmaf(__builtin_amdgcn_logf(d), wvln2,
                                fmaxf(u, 0.0f) * wv);
        // Fire-and-forget LDS atomics: single ds_add op each, no RMW
        // round trip, no s_wait_dscnt, private per-thread slots so
        // deterministic and conflict-free.
        atomicAdd(&sSum[b][tid], bcew);          // ds_add_f32 (no return)
        atomicAdd(&sCnt[b][tid], 1u);            // ds_add_u32 (no return)
      }
    }
    // top-of-next-iteration barrier protects buffer reuse
  }

  sWred[tid] = wacc * (float)NCLS;               // each row contributes 8*wv
  __syncthreads();                               // waits DScnt -> atomics done

  if (tid < NBINS) {
    float s = 0.0f; unsigned c = 0u;
    for (int i = 0; i < BLOCK; ++i) { s += sSum[tid][i]; c += sCnt[tid][i]; }
    atomicAdd(&gCnt[tid], c);                    // exact integer counts
    atomicAdd(&gSum[tid], s);
  }
  if (tid == 0) {
    float w = 0.0f;
    for (int i = 0; i < BLOCK; ++i) w += sWred[i];
    atomicAdd(gW, w);
  }
}

__global__ void ghm_final(const unsigned* __restrict__ gCnt,
                          const float* __restrict__ gSum,
                          const float* __restrict__ gW,
                          float* __restrict__ out) {
  if (threadIdx.x == 0 && blockIdx.x == 0) {
    const float tot = (float)NROWS * (float)NCLS;   // 33554432, exact in fp32
    float nz = 0.0f, acc = 0.0f;
    for (int b = 0; b < NBINS; ++b) {
      const unsigned c = gCnt[b];
      if (c > 0u) { nz += 1.0f; acc += gSum[b] * (tot / (float)c); }
    }
    out[0] = acc / fmaxf(nz, 1.0f) / gW[0];
  }
}

extern "C" void kernel_launch(void* const* d_in, const int* in_sizes, int n_in,
                              void* d_out, int out_size, void* d_ws, size_t ws_size,
                              hipStream_t stream) {
  const float* pred   = (const float*)d_in[0];
  const int*   target = (const int*)d_in[1];
  const float* weight = (const float*)d_in[2];
  // d_in[3] is `bins` == 10, compile-time constant here.

  unsigned* gCnt = (unsigned*)d_ws;
  float*    gSum = (float*)((char*)d_ws + 64);
  float*    gW   = (float*)((char*)d_ws + 128);

  ghm_zero<<<1, 64, 0, stream>>>(gCnt, gSum, gW);
  ghm_pass1<<<NBLK, BLOCK, 0, stream>>>(pred, target, weight, gCnt, gSum, gW);
  ghm_final<<<1, 1, 0, stream>>>(gCnt, gSum, gW, (float*)d_out);
}